// Loss_39548058861680
// MI455X (gfx1250) — compile-verified
//
#include <hip/hip_runtime.h>

// Clang vector types (HIP's float4 is a class type and would be rejected by
// __builtin_nontemporal_load; WMMA builtins want ext vectors).
typedef __attribute__((ext_vector_type(4))) float f32x4;
typedef __attribute__((ext_vector_type(2))) float v2f;
typedef __attribute__((ext_vector_type(8))) float v8f;

// Problem constants (from reference): B=64, K=16, C=16, H=256, W=256
#define N_HM   67108864    // B*C*H*W
#define N_PAIR 1024        // B*K
#define N_REG  2048        // B*K*2

__device__ __forceinline__ float sigmoid_clamped(float x) {
    // v_exp_f32 + v_rcp_f32 (1-ULP) instead of the IEEE v_div_* sequence:
    // keeps per-element VALU cost well under the HBM stream time.
    float s = __builtin_amdgcn_rcpf(1.0f + __expf(-x));
    return fminf(fmaxf(s, 1.0e-4f), 1.0f - 1.0e-4f);
}

// ---------------------------------------------------------------------------
// Kernel 1: streaming MSE over hm vs next_belief_maps.
// 537 MB single-pass @ 23.3 TB/s ~= 23 us floor. NT b128 loads (data > 192 MB
// L2, read once), global_prefetch_b8 one stride ahead, 32 B/thread/iter.
// ---------------------------------------------------------------------------
__global__ void hm_mse_kernel(const float* __restrict__ hm,
                              const float* __restrict__ nbm,
                              float* __restrict__ acc, int n) {
    const f32x4* hm4 = (const f32x4*)hm;
    const f32x4* nb4 = (const f32x4*)nbm;
    const int stride = gridDim.x * blockDim.x;
    const int tid    = blockIdx.x * blockDim.x + threadIdx.x;
    const int n8     = n >> 3;                       // # of 8-float chunks

    float acc0 = 0.0f, acc1 = 0.0f;

    for (int i = tid; i < n8; i += stride) {
        if (i + stride < n8) {                       // global_prefetch_b8
            __builtin_prefetch(&hm4[2 * (i + stride)], 0, 1);
            __builtin_prefetch(&nb4[2 * (i + stride)], 0, 1);
        }
        f32x4 h0 = __builtin_nontemporal_load(&hm4[2 * i + 0]);
        f32x4 h1 = __builtin_nontemporal_load(&hm4[2 * i + 1]);
        f32x4 t0 = __builtin_nontemporal_load(&nb4[2 * i + 0]);
        f32x4 t1 = __builtin_nontemporal_load(&nb4[2 * i + 1]);
#pragma unroll
        for (int c = 0; c < 4; ++c) {
            float d0 = sigmoid_clamped(h0[c]) - t0[c];
            float d1 = sigmoid_clamped(h1[c]) - t1[c];
            acc0 = fmaf(d0, d0, acc0);
            acc1 = fmaf(d1, d1, acc1);
        }
    }
    // scalar tail (n not divisible by 8)
    for (int i = (n8 << 3) + tid; i < n; i += stride) {
        float d = sigmoid_clamped(hm[i]) - nbm[i];
        acc0 = fmaf(d, d, acc0);
    }

    float local = acc0 + acc1;
    // wave32 tree reduction
#pragma unroll
    for (int off = 16; off > 0; off >>= 1)
        local += __shfl_xor(local, off, 32);

    __shared__ float red[8];
    const int lane = threadIdx.x & 31;
    const int wave = threadIdx.x >> 5;
    if (lane == 0) red[wave] = local;
    __syncthreads();
    if (threadIdx.x == 0) {
        float s = 0.0f;
#pragma unroll
        for (int w = 0; w < 8; ++w) s += red[w];
        atomicAdd(acc, s);  // one global_atomic_add_f32 per block
    }
}

// ---------------------------------------------------------------------------
// Kernel 2: gather smooth-L1 (reg term), then finalize the scalar loss.
// Block reduction of the 256 partials goes through V_WMMA_F32_16X16X4_F32:
//   A(16x4) = 64 folded partials, B(4x16) = ones, C = 0
//   => D[m][n] = rowsum[m]; lane0's 8 acc VGPRs hold rowsum[0..7], lane16's
//      hold rowsum[8..15]; one shfl_xor(16) completes the exact f32 sum.
// EXEC is all-ones within wave 0 (wave-uniform branch), per WMMA rules.
// ---------------------------------------------------------------------------
__global__ void reg_finalize_kernel(const float* __restrict__ reg,
                                    const float* __restrict__ reg_target,
                                    const int*  __restrict__ cord,
                                    const float* __restrict__ acc,
                                    float* __restrict__ out) {
    __shared__ float part[256];
    float local = 0.0f;

    for (int i = threadIdx.x; i < N_PAIR; i += 256) {
        const int b = i >> 4;                       // K = 16
        const int x = cord[i * 2 + 0];
        const int y = cord[i * 2 + 1];
        // reg layout [B,2,H,W]: ((b*2+c)*H + y)*W + x
        const size_t base = (size_t)b * 131072 + (size_t)y * 256 + (size_t)x;
        const float v0 = reg[base];
        const float v1 = reg[base + 65536];
        const float t0 = reg_target[i * 2 + 0];
        const float t1 = reg_target[i * 2 + 1];
        const float d0 = fabsf(v0 - t0);
        const float d1 = fabsf(v1 - t1);
        local += (d0 < 1.0f ? 0.5f * d0 * d0 : d0 - 0.5f);
        local += (d1 < 1.0f ? 0.5f * d1 * d1 : d1 - 0.5f);
    }
    part[threadIdx.x] = local;
    __syncthreads();

    if (threadIdx.x < 32) {                         // wave 0, EXEC = all ones
        const int lane = threadIdx.x;
        const int m  = lane & 15;
        const int kb = (lane < 16) ? 0 : 2;         // ISA A-matrix 16x4 layout
        const int i0 = m * 4 + kb;
        float ax = 0.0f, ay = 0.0f;
#pragma unroll
        for (int j = 0; j < 4; ++j) {               // fold 256 -> 64 values
            ax += part[i0 + 0 + 64 * j];
            ay += part[i0 + 1 + 64 * j];
        }
        v2f a    = {ax, ay};
        v2f ones = {1.0f, 1.0f};
        v8f c    = {};
        v8f d = __builtin_amdgcn_wmma_f32_16x16x4_f32(
            /*neg_a=*/false, a, /*neg_b=*/false, ones,
            /*c_mod=*/(short)0, c, /*reuse_a=*/false, /*reuse_b=*/false);

        float t = d[0] + d[1] + d[2] + d[3] + d[4] + d[5] + d[6] + d[7];
        t += __shfl_xor(t, 16, 32);                 // rowsums[0..7]+[8..15]

        if (lane == 0) {
            const float loss_reg = t * (1.0f / (float)N_REG);
            const float loss_hm  = acc[0] * (1.0f / (float)N_HM);
            // tracking weight is 0.0 and the term is finite -> exactly 0
            out[0] = loss_hm + 0.01f * loss_reg;
        }
    }
}

// ---------------------------------------------------------------------------
extern "C" void kernel_launch(void* const* d_in, const int* in_sizes, int n_in,
                              void* d_out, int out_size, void* d_ws, size_t ws_size,
                              hipStream_t stream) {
    const float* hm         = (const float*)d_in[0];
    const float* reg        = (const float*)d_in[1];
    // d_in[2] = tracking        (unused: weight 0.0)
    const float* nbm        = (const float*)d_in[3];
    const float* reg_target = (const float*)d_in[4];
    // d_in[5] = tracking_target (unused: weight 0.0)
    const int*   cord       = (const int*)d_in[6];

    float* acc = (float*)d_ws;
    hipMemsetAsync(acc, 0, sizeof(float), stream);  // graph-capturable

    const int n = in_sizes[0];                      // 67,108,864 f32 elements
    hm_mse_kernel<<<4096, 256, 0, stream>>>(hm, nbm, acc, n);
    reg_finalize_kernel<<<1, 256, 0, stream>>>(reg, reg_target, cord, acc,
                                               (float*)d_out);
}